// AGDNConv_14173392077061
// MI455X (gfx1250) — compile-verified
//
#include <hip/hip_runtime.h>
#include <hip/hip_bf16.h>

#define NN   100000
#define EE   1600000
#define INF_ 128
#define HH   3
#define DD   16
#define HD   48          // HH*DD
#define KK   3
#define NH   (NN*HH)     // 300000
#define EH   (EE*HH)     // 4800000
#define NHD  (NN*HD)     // 4800000
#define NEG_SLOPE 0.2f
#define EPSF 1e-9f

typedef __attribute__((ext_vector_type(2))) float v2f;
typedef __attribute__((ext_vector_type(8))) float v8f;

__device__ __forceinline__ float leaky_(float x) { return x >= 0.f ? x : NEG_SLOPE * x; }

// monotonic float -> uint key for atomicMax over signed floats
__device__ __forceinline__ unsigned fkey(float x) {
    unsigned u = __float_as_uint(x);
    return (u & 0x80000000u) ? ~u : (u | 0x80000000u);
}
__device__ __forceinline__ float fdec(unsigned u) {
    return (u & 0x80000000u) ? __uint_as_float(u ^ 0x80000000u) : __uint_as_float(~u);
}

// ---------------------------------------------------------------------------
// GEMM: ft[N,48] = feat[N,128] @ W[128,48] using V_WMMA_F32_16X16X4_F32.
// One wave per 16x16 output tile. 6250 M-tiles x 3 N-tiles = 18750 tiles.
// ---------------------------------------------------------------------------
__global__ __launch_bounds__(256) void k_gemm(const float* __restrict__ A,
                                              const float* __restrict__ W,
                                              float* __restrict__ C) {
    const int lane = threadIdx.x & 31;
    const int wave = threadIdx.x >> 5;
    const int tile = blockIdx.x * 8 + wave;
    const int NTILES = (NN / 16) * 3;
    if (tile >= NTILES) return;              // wave-uniform exit (EXEC stays all-1)

    const int mt   = tile / 3;
    const int nt   = tile - mt * 3;
    const int half = lane >> 4;              // 0: K{0,1}, 1: K{2,3}
    const int l    = lane & 15;

    const float* Arow = A + (size_t)(mt * 16 + l) * INF_;
    const int ncol    = nt * 16 + l;

    v8f acc = {};
    #pragma unroll 4
    for (int k0 = 0; k0 < INF_; k0 += 4) {
        const int ka = k0 + half * 2;
        v2f a = *(const v2f*)(Arow + ka);    // A[m][ka], A[m][ka+1]
        v2f b;
        b.x = W[ka * HD + ncol];             // B[ka][n]
        b.y = W[(ka + 1) * HD + ncol];       // B[ka+1][n]
        acc = __builtin_amdgcn_wmma_f32_16x16x4_f32(false, a, false, b,
                                                    (short)0, acc, false, false);
    }
    // C layout: VGPR i -> row (half*8 + i), column = lane&15
    float* Cb = C + (size_t)(mt * 16 + half * 8) * HD + nt * 16 + l;
    #pragma unroll
    for (int i = 0; i < 8; ++i) Cb[i * HD] = acc[i];
}

// el/er: per-(node,head) attention projections
__global__ void k_node_scores(const float* __restrict__ ft,
                              const float* __restrict__ attn_l,
                              const float* __restrict__ attn_r,
                              float* __restrict__ el, float* __restrict__ er) {
    int t = blockIdx.x * blockDim.x + threadIdx.x;
    if (t >= NH) return;
    int n = t / 3, h = t - n * 3;
    const float* f = ft + (size_t)n * HD + h * DD;
    float sl = 0.f, sr = 0.f;
    #pragma unroll
    for (int d = 0; d < DD; ++d) {
        sl += f[d] * attn_l[h * DD + d];
        sr += f[d] * attn_r[h * DD + d];
    }
    el[t] = sl;
    er[t] = sr;
}

__global__ void k_fill0(float* __restrict__ p, int n) {
    int i = blockIdx.x * blockDim.x + threadIdx.x;
    if (i < n) p[i] = 0.f;
}

// pass 1: edge scores + segment max (dst and src)
__global__ void k_edge_score(const int* __restrict__ src, const int* __restrict__ dst,
                             const float* __restrict__ el, const float* __restrict__ er,
                             float* __restrict__ esc,
                             unsigned* __restrict__ maxd, unsigned* __restrict__ maxs) {
    int i = blockIdx.x * blockDim.x + threadIdx.x;
    if (i >= EH) return;
    int e = i / 3, h = i - e * 3;
    int s = src[e], d = dst[e];
    float x = leaky_(el[s * 3 + h] + er[d * 3 + h]);
    esc[i] = x;
    unsigned k = fkey(x);
    atomicMax(&maxd[d * 3 + h], k);
    atomicMax(&maxs[s * 3 + h], k);
}

// pass 2: segment exp-sums
__global__ void k_edge_sum(const int* __restrict__ src, const int* __restrict__ dst,
                           const float* __restrict__ esc,
                           const unsigned* __restrict__ maxd, const unsigned* __restrict__ maxs,
                           float* __restrict__ sumd, float* __restrict__ sums) {
    int i = blockIdx.x * blockDim.x + threadIdx.x;
    if (i >= EH) return;
    int e = i / 3, h = i - e * 3;
    int s = src[e], d = dst[e];
    float x = esc[i];
    atomicAdd(&sumd[d * 3 + h], expf(x - fdec(maxd[d * 3 + h])));
    atomicAdd(&sums[s * 3 + h], expf(x - fdec(maxs[s * 3 + h])));
}

// pass 3: symmetric attention a = sqrt(clip(softmax_dst)*clip(softmax_src))
__global__ void k_edge_attn(const int* __restrict__ src, const int* __restrict__ dst,
                            const float* __restrict__ esc,
                            const unsigned* __restrict__ maxd, const unsigned* __restrict__ maxs,
                            const float* __restrict__ sumd, const float* __restrict__ sums,
                            float* __restrict__ attn) {
    int i = blockIdx.x * blockDim.x + threadIdx.x;
    if (i >= EH) return;
    int e = i / 3, h = i - e * 3;
    int s = src[e], d = dst[e];
    float x = esc[i];
    float ad = fmaxf(expf(x - fdec(maxd[d * 3 + h])) / sumd[d * 3 + h], EPSF);
    float as = fmaxf(expf(x - fdec(maxs[s * 3 + h])) / sums[s * 3 + h], EPSF);
    attn[i] = sqrtf(ad * as);
}

// per-hop normalization + pos-emb; also emits dot with a hop-attention vector
__global__ void k_feat_trans(const float* __restrict__ cur, int hop,
                             const float* __restrict__ scales, const float* __restrict__ offs,
                             const float* __restrict__ pos_emb, const float* __restrict__ hattn,
                             float* __restrict__ hout, float* __restrict__ adot) {
    int t = blockIdx.x * blockDim.x + threadIdx.x;
    if (t >= NH) return;
    int n = t / 3, h = t - n * 3;
    const float* x = cur + (size_t)n * HD + h * DD;
    float v[DD], s = 0.f, s2 = 0.f;
    #pragma unroll
    for (int d = 0; d < DD; ++d) { float y = x[d]; v[d] = y; s += y; s2 += y * y; }
    float mean = s * (1.0f / DD);
    float var  = s2 * (1.0f / DD) - mean * mean + EPSF;
    float r    = rsqrtf(var);
    const float* sc = scales + hop * HD + h * DD;
    const float* of = offs   + hop * HD + h * DD;
    const float* pe = pos_emb + h * ((KK + 1) * DD) + hop * DD;
    float dot = 0.f;
    #pragma unroll
    for (int d = 0; d < DD; ++d) {
        float y = (v[d] - mean) * sc[d] * r + of[d] + pe[d];
        if (hout) hout[(size_t)n * HD + h * DD + d] = y;
        dot += y * hattn[h * DD + d];
    }
    adot[t] = dot;
}

// one diffusion hop: out[dst] += a * cur[src]   (per edge, per head, D=16)
__global__ void k_propagate(const int* __restrict__ src, const int* __restrict__ dst,
                            const float* __restrict__ attn,
                            const float* __restrict__ cur, float* __restrict__ out) {
    int i = blockIdx.x * blockDim.x + threadIdx.x;
    if (i >= EH) return;
    int e = i / 3, h = i - e * 3;
    float a = attn[i];
    const float4* cs = (const float4*)(cur + (size_t)src[e] * HD + h * DD);
    float* od = out + (size_t)dst[e] * HD + h * DD;
    float4 c0 = cs[0], c1 = cs[1], c2 = cs[2], c3 = cs[3];
    atomicAdd(od + 0,  a * c0.x); atomicAdd(od + 1,  a * c0.y);
    atomicAdd(od + 2,  a * c0.z); atomicAdd(od + 3,  a * c0.w);
    atomicAdd(od + 4,  a * c1.x); atomicAdd(od + 5,  a * c1.y);
    atomicAdd(od + 6,  a * c1.z); atomicAdd(od + 7,  a * c1.w);
    atomicAdd(od + 8,  a * c2.x); atomicAdd(od + 9,  a * c2.y);
    atomicAdd(od + 10, a * c2.z); atomicAdd(od + 11, a * c2.w);
    atomicAdd(od + 12, a * c3.x); atomicAdd(od + 13, a * c3.y);
    atomicAdd(od + 14, a * c3.z); atomicAdd(od + 15, a * c3.w);
}

// hop-attention softmax over K=3 + weighted sum + bias
__global__ void k_final(const float* __restrict__ hst, const float* __restrict__ ar,
                        const float* __restrict__ al, const float* __restrict__ bias,
                        float* __restrict__ out) {
    int t = blockIdx.x * blockDim.x + threadIdx.x;
    if (t >= NH) return;
    int n = t / 3, h = t - n * 3;
    float q  = al[t];
    float a0 = leaky_(ar[0 * NH + t] + q);
    float a1 = leaky_(ar[1 * NH + t] + q);
    float a2 = leaky_(ar[2 * NH + t] + q);
    float m  = fmaxf(a0, fmaxf(a1, a2));
    float w0 = expf(a0 - m), w1 = expf(a1 - m), w2 = expf(a2 - m);
    float inv = 1.f / (w0 + w1 + w2);
    w0 *= inv; w1 *= inv; w2 *= inv;
    const size_t base = (size_t)n * HD + h * DD;
    const float* h0 = hst + 0 * (size_t)NHD + base;
    const float* h1 = hst + 1 * (size_t)NHD + base;
    const float* h2 = hst + 2 * (size_t)NHD + base;
    #pragma unroll
    for (int d = 0; d < DD; ++d)
        out[base + d] = bias[h * DD + d] + w0 * h0[d] + w1 * h1[d] + w2 * h2[d];
}

extern "C" void kernel_launch(void* const* d_in, const int* in_sizes, int n_in,
                              void* d_out, int out_size, void* d_ws, size_t ws_size,
                              hipStream_t stream) {
    (void)in_sizes; (void)n_in; (void)out_size;
    const float* feat    = (const float*)d_in[0];
    const int*   src     = (const int*)d_in[1];
    const int*   dst     = (const int*)d_in[2];
    const float* fcW     = (const float*)d_in[3];
    const float* attn_l  = (const float*)d_in[4];
    const float* attn_r  = (const float*)d_in[5];
    const float* hop_l   = (const float*)d_in[6];
    const float* hop_r   = (const float*)d_in[7];
    const float* pos_emb = (const float*)d_in[8];
    const float* scales  = (const float*)d_in[9];
    const float* offs    = (const float*)d_in[10];
    const float* bias    = (const float*)d_in[11];
    float* out = (float*)d_out;

    float* ws = (float*)d_ws;
    size_t off = 0;
    auto alloc = [&](size_t n) { float* p = ws + off; off += n; return p; };
    float*    ft   = alloc(NHD);
    float*    esc  = alloc(EH);
    float*    attn = alloc(EH);
    float*    curA = alloc(NHD);
    float*    curB = alloc(NHD);
    float*    hst  = alloc(3 * (size_t)NHD);
    float*    el   = alloc(NH);
    float*    er   = alloc(NH);
    unsigned* maxd = (unsigned*)alloc(NH);
    unsigned* maxs = (unsigned*)alloc(NH);   // maxd,maxs,sumd,sums contiguous
    float*    sumd = alloc(NH);
    float*    sums = alloc(NH);
    float*    al   = alloc(NH);
    float*    ar   = alloc(3 * (size_t)NH);
    if (ws_size < off * sizeof(float)) return;

    const dim3 B(256);
    const int gNH  = (NH  + 255) / 256;
    const int gEH  = (EH  + 255) / 256;
    const int gNHD = (NHD + 255) / 256;

    k_gemm<<<dim3((18750 + 7) / 8), B, 0, stream>>>(feat, fcW, ft);
    k_node_scores<<<dim3(gNH), B, 0, stream>>>(ft, attn_l, attn_r, el, er);
    k_fill0<<<dim3((4 * NH + 255) / 256), B, 0, stream>>>((float*)maxd, 4 * NH);
    k_edge_score<<<dim3(gEH), B, 0, stream>>>(src, dst, el, er, esc, maxd, maxs);
    k_edge_sum<<<dim3(gEH), B, 0, stream>>>(src, dst, esc, maxd, maxs, sumd, sums);
    k_edge_attn<<<dim3(gEH), B, 0, stream>>>(src, dst, esc, maxd, maxs, sumd, sums, attn);

    // hop 0 query: only its hop_attn_l projection is needed downstream
    k_feat_trans<<<dim3(gNH), B, 0, stream>>>(ft, 0, scales, offs, pos_emb, hop_l,
                                              (float*)nullptr, al);

    const float* cur = ft;
    float* nxt = curA;
    for (int k = 1; k <= KK; ++k) {
        k_fill0<<<dim3(gNHD), B, 0, stream>>>(nxt, NHD);
        k_propagate<<<dim3(gEH), B, 0, stream>>>(src, dst, attn, cur, nxt);
        k_feat_trans<<<dim3(gNH), B, 0, stream>>>(nxt, k, scales, offs, pos_emb, hop_r,
                                                  hst + (size_t)(k - 1) * NHD,
                                                  ar + (size_t)(k - 1) * NH);
        cur = nxt;
        nxt = (k == 1) ? curB : curA;   // ping-pong raw hop buffers
    }

    k_final<<<dim3(gNH), B, 0, stream>>>(hst, ar, al, bias, out);
}